// DifferentialEKVConv2d_12154757447728
// MI455X (gfx1250) — compile-verified
//
#include <hip/hip_runtime.h>

typedef __attribute__((ext_vector_type(2))) float v2f;
typedef __attribute__((ext_vector_type(8))) float v8f;

#define HW     32
#define C_IN   16
#define O_CH   32
#define RTOT   144          // C_IN * 3 * 3
#define X_CHUNKS  512       // 16c * 4rows * 32cols * 4B / 16B
#define T_CHUNKS  1152      // 32 * 144 * 4B / 16B

// out[n,o,h,w] = TIA * sum_r [ f(x_patch - tpos) - f(x_patch - tneg) ]
// f(u) = ALPHA * ( sp(u/PHI)^2 - sp(u/PHI - VD/PHI)^2 ),  sp = softplus
// Using base-2 HW trans ops:  sp(v) = ln2 * log2(1 + 2^(v*log2e))
// and  2^((v-d)*log2e) = 2^(v*log2e) * e^(-d)  -> 1 exp + 2 log per threshold.
__global__ __launch_bounds__(256) void ekv_diff_conv(
    const float* __restrict__ x,
    const float* __restrict__ theta_pos,
    const float* __restrict__ theta_neg,
    float* __restrict__ out)
{
    __shared__ __align__(16) float xs[C_IN * 4 * HW];   //  8 KB: [c][rr][col], rr = row-(h0-1)
    __shared__ __align__(16) float tp[O_CH * RTOT];     // 18 KB: [o][r]
    __shared__ __align__(16) float tn[O_CH * RTOT];     // 18 KB

    const int tid = threadIdx.x;
    const int blk = blockIdx.x;          // 256 blocks = 16 images * 16 row-pairs
    const int n   = blk >> 4;
    const int h0  = (blk & 15) << 1;     // this block covers output rows h0, h0+1

    // LDS byte offsets (low 32 bits of a generic pointer to LDS == LDS offset)
    const unsigned lds_x  = (unsigned)(unsigned long long)(void*)&xs[0];
    const unsigned lds_tp = (unsigned)(unsigned long long)(void*)&tp[0];
    const unsigned lds_tn = (unsigned)(unsigned long long)(void*)&tn[0];

    // ---------- CDNA5 async global->LDS staging ----------
    for (int i = tid; i < T_CHUNKS; i += 256) {
        unsigned long long gp = (unsigned long long)theta_pos + (unsigned long long)i * 16u;
        unsigned long long gn = (unsigned long long)theta_neg + (unsigned long long)i * 16u;
        unsigned lp  = lds_tp + (unsigned)i * 16u;
        unsigned ln_ = lds_tn + (unsigned)i * 16u;
        asm volatile("global_load_async_to_lds_b128 %0, %1, off" :: "v"(lp),  "v"(gp) : "memory");
        asm volatile("global_load_async_to_lds_b128 %0, %1, off" :: "v"(ln_), "v"(gn) : "memory");
    }
    for (int i = tid; i < X_CHUNKS; i += 256) {
        int c  = i >> 5;            // input channel
        int rr = (i >> 3) & 3;      // staged row slot 0..3  (input row h0-1+rr)
        int cg = i & 7;             // 16B column group
        int row = h0 - 1 + rr;
        if (row >= 0 && row < HW) { // out-of-range rows: never read (masked), skip copy
            unsigned long long gx = (unsigned long long)x +
                ((unsigned long long)(((n * C_IN + c) * HW + row) * HW + cg * 4)) * 4ull;
            unsigned lx = lds_x + (unsigned)i * 16u;
            asm volatile("global_load_async_to_lds_b128 %0, %1, off" :: "v"(lx), "v"(gx) : "memory");
        }
    }
    asm volatile("s_wait_asynccnt 0" ::: "memory");
    __syncthreads();

    // ---------- per-wave 16-pixel x 16-ochannel tile ----------
    const int lane  = tid & 31;
    const int wave  = tid >> 5;          // 8 waves
    const int seg   = wave & 3;          // 4 pixel segments: 2 rows x 2 half-rows
    const int otile = wave >> 2;         // 2 output-channel tiles
    const int h  = h0 + (seg >> 1);
    const int w0 = (seg & 1) << 4;
    const int o0 = otile << 4;
    const int l15 = lane & 15;
    const bool lolane = lane < 16;

    const float invPHI = 1.0f / (2.0f * 1.5f * 0.026f);   // 1/PHI
    const float L2E    = 1.4426950408889634f;             // log2(e)
    const float dshift = 0.2f * invPHI;                   // VD/PHI
    const float csh    = __builtin_amdgcn_exp2f(-dshift * L2E);  // e^{-VD/PHI}
    const float a1     = lolane ? 1.0f : 0.0f;            // A "ones" column (K=1); hi lanes hold K=2,3 = 0

    v8f acc  = {0.f,0.f,0.f,0.f,0.f,0.f,0.f,0.f};
    const v8f zero = {0.f,0.f,0.f,0.f,0.f,0.f,0.f,0.f};

    for (int c = 0; c < C_IN; ++c) {
        for (int dh = 0; dh < 3; ++dh) {
            const int row = h + dh - 1;
            const bool rvalid = (row >= 0) && (row < HW);
            int rr = row - (h0 - 1);
            rr = rr < 0 ? 0 : (rr > 3 ? 3 : rr);
            #pragma unroll
            for (int dw = 0; dw < 3; ++dw) {
                const int r = c * 9 + dh * 3 + dw;
                // A-matrix (16x4 f32): lanes 0-15 = {x[M]*invPHI, 1}; lanes 16-31 = {0,0}
                int col = w0 + l15 + dw - 1;
                const bool cvalid = (col >= 0) && (col < HW);
                col = col < 0 ? 0 : (col > HW - 1 ? HW - 1 : col);
                const float xv = (lolane && rvalid && cvalid) ? xs[(c * 4 + rr) * HW + col] : 0.0f;
                // B-matrix (4x16 f32): lanes 0-15 = {1, -t[N]*invPHI}; lanes 16-31 = {0,0}
                const float tpv = lolane ? tp[(o0 + l15) * RTOT + r] : 0.0f;
                const float tnv = lolane ? tn[(o0 + l15) * RTOT + r] : 0.0f;
                v2f A;  A[0]  = xv * invPHI;  A[1]  = a1;
                v2f Bp; Bp[0] = a1;           Bp[1] = -tpv * invPHI;
                v2f Bn; Bn[0] = a1;           Bn[1] = -tnv * invPHI;
                // D[p,o] = x[p]/PHI - t[o]/PHI  via rank-2 WMMA
                v8f dp = __builtin_amdgcn_wmma_f32_16x16x4_f32(
                    false, A, false, Bp, (short)0, zero, false, false);
                v8f dn = __builtin_amdgcn_wmma_f32_16x16x4_f32(
                    false, A, false, Bn, (short)0, zero, false, false);
                #pragma unroll
                for (int j = 0; j < 8; ++j) {
                    const float up = dp[j];
                    const float zp = __builtin_amdgcn_exp2f(up * L2E);
                    const float ap = __builtin_amdgcn_logf(1.0f + zp);
                    const float bp = __builtin_amdgcn_logf(1.0f + zp * csh);
                    const float un = dn[j];
                    const float zn = __builtin_amdgcn_exp2f(un * L2E);
                    const float an = __builtin_amdgcn_logf(1.0f + zn);
                    const float bn = __builtin_amdgcn_logf(1.0f + zn * csh);
                    acc[j] += (ap * ap - bp * bp) - (an * an - bn * bn);
                }
            }
        }
    }

    // out = acc * ALPHA * TIA * ln2^2   (softplus = ln2 * log2(...))
    const float SCALE = 1e-5f * 2000.0f * 0.6931471805599453f * 0.6931471805599453f;
    const int o = o0 + l15;
    const int pixbase = lolane ? 0 : 8;      // C/D layout: lanes>=16 hold M = vgpr+8
    #pragma unroll
    for (int j = 0; j < 8; ++j) {
        const int w = w0 + pixbase + j;
        out[(((n * O_CH) + o) * HW + h) * HW + w] = acc[j] * SCALE;
    }
}

extern "C" void kernel_launch(void* const* d_in, const int* in_sizes, int n_in,
                              void* d_out, int out_size, void* d_ws, size_t ws_size,
                              hipStream_t stream) {
    const float* x         = (const float*)d_in[0];
    const float* theta_pos = (const float*)d_in[1];
    const float* theta_neg = (const float*)d_in[2];
    float* out = (float*)d_out;
    (void)in_sizes; (void)n_in; (void)out_size; (void)d_ws; (void)ws_size;

    dim3 grid(16 * 16);   // 16 images * 16 row-pairs
    dim3 block(256);      // 8 wave32
    ekv_diff_conv<<<grid, block, 0, stream>>>(x, theta_pos, theta_neg, out);
}